// VulnerabilityGAT_7756710937191
// MI455X (gfx1250) — compile-verified
//
#include <hip/hip_runtime.h>

#define HEADS 8
#define CPH   32
#define HID   256
#define NEG_SLOPE 0.2f

typedef __attribute__((ext_vector_type(2))) float v2f;
typedef __attribute__((ext_vector_type(8))) float v8f;

__device__ __forceinline__ float leakyr(float x) { return x > 0.f ? x : NEG_SLOPE * x; }
__device__ __forceinline__ float eluf(float x)   { return x > 0.f ? x : (__expf(x) - 1.f); }

__device__ __forceinline__ void edge_nodes(const long long* __restrict__ ei, int E, int e,
                                           int& s, int& d) {
    if (e < E) { s = (int)ei[e]; d = (int)ei[E + e]; }
    else       { s = e - E;      d = e - E; }          // self loops appended after edges
}

// ---------------------------------------------------------------------------
// Out[M,256] = A[M,K] @ W[K,256]  via v_wmma_f32_16x16x4_f32 (fp32 all the way)
// 8 waves / block; wave w owns cols [w*32, w*32+32) of a 16-row strip.
// A frag (16x4 f32): lanes 0-15 -> M=lane, VGPR0=K0,VGPR1=K1; lanes 16-31 -> K2/K3.
// B frag (4x16 f32): mirrored (lanes 0-15: N=lane, K0/K1; lanes 16-31: K2/K3).
// C/D (16x16 f32): VGPR r -> M=r (lanes 0-15) / M=r+8 (lanes 16-31), N=lane&15.
// ---------------------------------------------------------------------------
__global__ __launch_bounds__(256) void gat_gemm(const float* __restrict__ A,
                                                const float* __restrict__ W,
                                                float* __restrict__ Out,
                                                int M, int K) {
    const int wave  = threadIdx.x >> 5;
    const int lane  = threadIdx.x & 31;
    const int l16   = lane & 15;
    const int khalf = (lane >> 4) << 1;          // 0 or 2
    const int row0  = blockIdx.x * 16;
    const int cb0   = wave * 32;

    int arow_idx = row0 + l16;
    if (arow_idx >= M) arow_idx = M - 1;         // clamp (M is a multiple of 16 here)
    const float* __restrict__ arow = A + (size_t)arow_idx * K;

    v8f c0 = {};
    v8f c1 = {};
    for (int k0 = 0; k0 < K; k0 += 4) {
        v2f a, b0, b1;
        a[0] = arow[k0 + khalf];
        a[1] = arow[k0 + khalf + 1];
        const float* __restrict__ w0 = W + (size_t)(k0 + khalf) * HID;
        b0[0] = w0[cb0 + l16];
        b0[1] = w0[HID + cb0 + l16];
        b1[0] = w0[cb0 + 16 + l16];
        b1[1] = w0[HID + cb0 + 16 + l16];
        c0 = __builtin_amdgcn_wmma_f32_16x16x4_f32(false, a, false, b0, (short)0, c0, false, false);
        c1 = __builtin_amdgcn_wmma_f32_16x16x4_f32(false, a, false, b1, (short)0, c1, false, false);
    }

    const int mhi = (lane >> 4) * 8;
    for (int r = 0; r < 8; ++r) {
        int row = row0 + r + mhi;
        if (row < M) {
            float* __restrict__ orow = Out + (size_t)row * HID;
            orow[cb0 + l16]      = c0[r];
            orow[cb0 + 16 + l16] = c1[r];
        }
    }
}

// Per-node attention logits: al_s[n,h] = sum_c h[n,h,c]*a_s[h,c] (same for dst)
__global__ void gat_att_logits(const float* __restrict__ h,
                               const float* __restrict__ a_s,
                               const float* __restrict__ a_d,
                               float* __restrict__ al_s,
                               float* __restrict__ al_d, int N) {
    int i = blockIdx.x * blockDim.x + threadIdx.x;   // node*8 + head
    if (i >= N * HEADS) return;
    int head = i & 7;
    size_t node = (size_t)(i >> 3);
    const float* __restrict__ hp = h + node * HID + head * CPH;
    const float* __restrict__ as = a_s + head * CPH;
    const float* __restrict__ ad = a_d + head * CPH;
    float ss = 0.f, sd = 0.f;
#pragma unroll
    for (int c = 0; c < CPH; ++c) { float v = hp[c]; ss += v * as[c]; sd += v * ad[c]; }
    al_s[i] = ss;
    al_d[i] = sd;
}

// Init per layer: acc = broadcast bias, m = -inf, denom = 0
__global__ void gat_init(float* __restrict__ acc, float* __restrict__ m,
                         float* __restrict__ denom, const float* __restrict__ b, int N) {
    size_t i = (size_t)blockIdx.x * blockDim.x + threadIdx.x;
    if (i >= (size_t)N * HID) return;
    int c = (int)(i & 255);
    acc[i] = b[c];
    if (c < 8) {
        size_t n = i >> 8;
        m[n * 8 + c]     = -3.0e38f;
        denom[n * 8 + c] = 0.f;
    }
}

// Segment max of leaky-relu edge scores into m[dst,h]
__global__ void gat_edge_max(const float* __restrict__ al_s, const float* __restrict__ al_d,
                             const long long* __restrict__ ei, int E, int Etot,
                             float* __restrict__ m) {
    int e = blockIdx.x * blockDim.x + threadIdx.x;
    if (e >= Etot) return;
    int s, d; edge_nodes(ei, E, e, s, d);
#pragma unroll
    for (int h = 0; h < HEADS; ++h) {
        float v = leakyr(al_s[s * 8 + h] + al_d[d * 8 + h]);
        atomicMax(&m[d * 8 + h], v);
    }
}

// ex = exp(score - m[dst]); denom[dst] += ex
__global__ void gat_edge_exp(const float* __restrict__ al_s, const float* __restrict__ al_d,
                             const long long* __restrict__ ei, int E, int Etot,
                             const float* __restrict__ m, float* __restrict__ denom,
                             float* __restrict__ ex) {
    int e = blockIdx.x * blockDim.x + threadIdx.x;
    if (e >= Etot) return;
    int s, d; edge_nodes(ei, E, e, s, d);
#pragma unroll
    for (int h = 0; h < HEADS; ++h) {
        float v = leakyr(al_s[s * 8 + h] + al_d[d * 8 + h]);
        float x = __expf(v - m[d * 8 + h]);
        ex[(size_t)e * 8 + h] = x;
        atomicAdd(&denom[d * 8 + h], x);
    }
}

// Weighted scatter-add: one wave32 per edge, lane owns 8 contiguous channels
__global__ __launch_bounds__(256) void gat_aggregate(const float* __restrict__ hproj,
                                                     const long long* __restrict__ ei,
                                                     const float* __restrict__ ex,
                                                     const float* __restrict__ denom,
                                                     float* __restrict__ acc,
                                                     int E, int Etot) {
    const int wave = threadIdx.x >> 5;
    const int lane = threadIdx.x & 31;
    int e = blockIdx.x * 8 + wave;
    if (e >= Etot) return;
    int s, d; edge_nodes(ei, E, e, s, d);
    const int head = lane >> 2;                       // 8 channels/lane -> head = lane/4
    float alpha = ex[(size_t)e * 8 + head] / (denom[d * 8 + head] + 1e-16f);
    const float4* __restrict__ hp = (const float4*)(hproj + (size_t)s * HID + lane * 8);
    float4 v0 = hp[0];
    float4 v1 = hp[1];
    float* __restrict__ ap = acc + (size_t)d * HID + lane * 8;
    atomicAdd(ap + 0, v0.x * alpha);
    atomicAdd(ap + 1, v0.y * alpha);
    atomicAdd(ap + 2, v0.z * alpha);
    atomicAdd(ap + 3, v0.w * alpha);
    atomicAdd(ap + 4, v1.x * alpha);
    atomicAdd(ap + 5, v1.y * alpha);
    atomicAdd(ap + 6, v1.z * alpha);
    atomicAdd(ap + 7, v1.w * alpha);
}

// out = elu(acc) (+ prev residual)
__global__ void gat_epilogue(const float* __restrict__ acc, const float* __restrict__ prev,
                             float* __restrict__ out, size_t total) {
    size_t i = (size_t)blockIdx.x * blockDim.x + threadIdx.x;
    if (i >= total) return;
    float v = eluf(acc[i]);
    if (prev) v += prev[i];
    out[i] = v;
}

__global__ void gat_zero256(float* __restrict__ p) { p[threadIdx.x] = 0.f; }

// Column sums of h3 (coalesced: thread t = column t, loop over node chunk)
__global__ __launch_bounds__(256) void gat_colsum(const float* __restrict__ h,
                                                  float* __restrict__ gsum,
                                                  int N, int chunk) {
    int t  = threadIdx.x;
    int n0 = blockIdx.x * chunk;
    int n1 = n0 + chunk; if (n1 > N) n1 = N;
    float s = 0.f;
    for (int n = n0; n < n1; ++n) s += h[(size_t)n * HID + t];
    atomicAdd(&gsum[t], s);
}

// mean -> 256x128 matvec -> relu -> 128x2 matvec, single block
__global__ __launch_bounds__(256) void gat_classifier(const float* __restrict__ gsum,
                                                      const float* __restrict__ Wc1,
                                                      const float* __restrict__ bc1,
                                                      const float* __restrict__ Wc2,
                                                      const float* __restrict__ bc2,
                                                      float* __restrict__ out, int N) {
    __shared__ float g[HID];
    __shared__ float r[128];
    int t = threadIdx.x;
    g[t] = gsum[t] / (float)N;
    __syncthreads();
    if (t < 128) {
        float acc = bc1[t];
        for (int k = 0; k < HID; ++k) acc += g[k] * Wc1[k * 128 + t];
        r[t] = acc > 0.f ? acc : 0.f;
    }
    __syncthreads();
    if (t < 2) {
        float acc = bc2[t];
        for (int k = 0; k < 128; ++k) acc += r[k] * Wc2[k * 2 + t];
        out[t] = acc;
    }
}

// ---------------------------------------------------------------------------
static void run_layer(hipStream_t stream, const float* in, int K,
                      const float* W, const float* as_, const float* ad_, const float* b,
                      const long long* ei, int E, int Etot, int N,
                      float* proj, float* acc, float* al_s, float* al_d,
                      float* mbuf, float* dbuf, float* exbuf,
                      const float* prev, float* out) {
    int gemm_blocks = (N + 15) / 16;
    gat_gemm<<<gemm_blocks, 256, 0, stream>>>(in, W, proj, N, K);
    gat_att_logits<<<(N * HEADS + 255) / 256, 256, 0, stream>>>(proj, as_, ad_, al_s, al_d, N);
    gat_init<<<(int)(((size_t)N * HID + 255) / 256), 256, 0, stream>>>(acc, mbuf, dbuf, b, N);
    gat_edge_max<<<(Etot + 255) / 256, 256, 0, stream>>>(al_s, al_d, ei, E, Etot, mbuf);
    gat_edge_exp<<<(Etot + 255) / 256, 256, 0, stream>>>(al_s, al_d, ei, E, Etot, mbuf, dbuf, exbuf);
    gat_aggregate<<<(Etot + 7) / 8, 256, 0, stream>>>(proj, ei, exbuf, dbuf, acc, E, Etot);
    gat_epilogue<<<(int)(((size_t)N * HID + 255) / 256), 256, 0, stream>>>(acc, prev, out, (size_t)N * HID);
}

extern "C" void kernel_launch(void* const* d_in, const int* in_sizes, int n_in,
                              void* d_out, int out_size, void* d_ws, size_t ws_size,
                              hipStream_t stream) {
    const float*     x   = (const float*)d_in[0];
    const long long* ei  = (const long long*)d_in[1];   // int64 [2,E]
    const float* W1  = (const float*)d_in[2];
    const float* a1s = (const float*)d_in[3];
    const float* a1d = (const float*)d_in[4];
    const float* b1  = (const float*)d_in[5];
    const float* W2  = (const float*)d_in[6];
    const float* a2s = (const float*)d_in[7];
    const float* a2d = (const float*)d_in[8];
    const float* b2  = (const float*)d_in[9];
    const float* W3  = (const float*)d_in[10];
    const float* a3s = (const float*)d_in[11];
    const float* a3d = (const float*)d_in[12];
    const float* b3  = (const float*)d_in[13];
    const float* Wc1 = (const float*)d_in[14];
    const float* bc1 = (const float*)d_in[15];
    const float* Wc2 = (const float*)d_in[16];
    const float* bc2 = (const float*)d_in[17];

    const int IN_DIM = 128;
    const int N    = in_sizes[0] / IN_DIM;    // 50000
    const int E    = in_sizes[1] / 2;         // 400000
    const int Etot = E + N;                   // + self loops

    // Workspace layout (floats)
    float* ws    = (float*)d_ws;
    float* proj  = ws;                               // N*256
    float* acc   = proj  + (size_t)N * HID;          // N*256
    float* hA    = acc   + (size_t)N * HID;          // N*256  (h1)
    float* hB    = hA    + (size_t)N * HID;          // N*256  (h2)
    float* al_s  = hB    + (size_t)N * HID;          // N*8
    float* al_d  = al_s  + (size_t)N * 8;            // N*8
    float* mbuf  = al_d  + (size_t)N * 8;            // N*8
    float* dbuf  = mbuf  + (size_t)N * 8;            // N*8
    float* exbuf = dbuf  + (size_t)N * 8;            // Etot*8
    float* gsum  = exbuf + (size_t)Etot * 8;         // 256

    float* logits = (float*)d_out;
    float* h3     = logits + 2;                      // [N,256] written in place

    // Layer 1: h1 = elu(GAT(x))
    run_layer(stream, x, IN_DIM, W1, a1s, a1d, b1, ei, E, Etot, N,
              proj, acc, al_s, al_d, mbuf, dbuf, exbuf, nullptr, hA);
    // Layer 2: h2 = elu(GAT(h1)) + h1
    run_layer(stream, hA, HID, W2, a2s, a2d, b2, ei, E, Etot, N,
              proj, acc, al_s, al_d, mbuf, dbuf, exbuf, hA, hB);
    // Layer 3: h3 = elu(GAT(h2)) + h2  -> directly into d_out
    run_layer(stream, hB, HID, W3, a3s, a3d, b3, ei, E, Etot, N,
              proj, acc, al_s, al_d, mbuf, dbuf, exbuf, hB, h3);

    // Mean pooling + classifier MLP
    gat_zero256<<<1, 256, 0, stream>>>(gsum);
    const int csblocks = 256;
    const int chunk = (N + csblocks - 1) / csblocks;
    gat_colsum<<<csblocks, 256, 0, stream>>>(h3, gsum, N, chunk);
    gat_classifier<<<1, 256, 0, stream>>>(gsum, Wc1, bc1, Wc2, bc2, logits, N);
}